// TensorProduct_77421080478251
// MI455X (gfx1250) — compile-verified
//
#include <hip/hip_runtime.h>

// CDNA5 / gfx1250, wave32.
// e3nn uvw tensor product (128x0e+128x1o) x (1x0e+1x1o) -> (128x0e+128x1o).
// 6 GEMMs of 16(z) x 128(w) x 128(u) per z-tile via V_WMMA_F32_16X16X4_F32.
// B fragments (weights) live in VGPRs for the whole block; LDS holds only a
// double-buffered raw x1 tile staged via GLOBAL_LOAD_ASYNC_TO_LDS (ASYNCcnt),
// falling back to sync load+ds_store if the async builtin is unavailable.

typedef __attribute__((ext_vector_type(2))) float v2f;
typedef __attribute__((ext_vector_type(4))) float v4f;
typedef __attribute__((ext_vector_type(8))) float v8f;
typedef int v4i __attribute__((vector_size(16)));   // matches builtin param type

#define PX1 516                   // x1 tile row pitch (floats): 4-bank lane stride
#define X1BUF (16 * PX1)          // 8256 floats per buffer
#define OFF_X2 (2 * X1BUF)        // 2 x 64 floats of x2 tile
#define SMEM_FLOATS (OFF_X2 + 2 * 64)   // 16640 floats = 66560 B

#define TPB 8                     // z-tiles (of 16 rows) per block
#define INV_SQRT3 0.57735026918962576f

#define WMMA(A, B, C) \
  __builtin_amdgcn_wmma_f32_16x16x4_f32(false, (A), false, (B), (short)0, (C), false, false)

#if __has_builtin(__builtin_amdgcn_global_load_async_to_lds_b128)
#define HAVE_ASYNC_LDS 1
typedef __attribute__((address_space(1))) v4i* gas_ptr;
typedef __attribute__((address_space(3))) v4i* lds_ptr;
#else
#define HAVE_ASYNC_LDS 0
#endif

static __device__ __forceinline__ v2f mk2(float a, float b) {
  v2f r; r[0] = a; r[1] = b; return r;
}

static __device__ __forceinline__ void wait_async_all() {
#if __has_builtin(__builtin_amdgcn_s_wait_asynccnt)
  __builtin_amdgcn_s_wait_asynccnt(0);
#else
  asm volatile("s_wait_asynccnt 0" ::: "memory");
#endif
}

__global__ __launch_bounds__(256, 1) __attribute__((amdgpu_waves_per_eu(2, 2)))
void tp_uvw_wmma_kernel(const float* __restrict__ x1, const float* __restrict__ x2,
                        const float* __restrict__ wgt, float* __restrict__ out,
                        int Z, int nTiles) {
  extern __shared__ float smem[];
  const int tid = threadIdx.x;
  const int lane = tid & 31;
  const int h = lane >> 4;        // selects K pair within a WMMA fragment
  const int nn = lane & 15;       // A-matrix row / B-matrix column within tile
  const int wv = tid >> 5;        // wave id -> 16-column output block
  const int colbase = wv * 16;

  // ---- hoist all B fragments into VGPRs (once per block, L2-resident) ----
  // B layout (f32 4x16): lane l, vgpr j holds B[2*(l>>4)+j, l&15] = W[u0+j, w]
  v2f B0[32], B1[32], B2[32], B3[32];
  {
    const float* gw = wgt + colbase + nn;
#pragma unroll
    for (int kk = 0; kk < 32; ++kk) {
      const float* g = gw + (4 * kk + 2 * h) * 128;
      B0[kk] = mk2(g[0],         g[128]);
      B1[kk] = mk2(g[16384],     g[16384 + 128]);
      B2[kk] = mk2(g[32768],     g[32768 + 128]);
      B3[kk] = mk2(g[49152],     g[49152 + 128]);
    }
  }

#if HAVE_ASYNC_LDS
  // ---- async staging: global -> LDS directly, tracked by ASYNCcnt ----
  auto stage_async = [&](int z0, int buf) {
    float* bn = smem + buf * X1BUF;
#pragma unroll
    for (int i = 0; i < 8; ++i) {
      int idx = tid + (i << 8);           // 2048 float4 slots (16 rows x 128)
      int m = idx >> 5, c = idx & 31;
      int zm = z0 + m; if (zm >= Z) zm = Z - 1;
      __builtin_amdgcn_global_load_async_to_lds_b128(
          (gas_ptr)(x1 + (size_t)zm * 512 + c * 4),
          (lds_ptr)(bn + m * PX1 + c * 4),
          0, 0);
    }
  };
#endif

  const int tile0 = blockIdx.x * TPB;
#if !HAVE_ASYNC_LDS
  v4f sreg[8];
#endif
  float x2v = 0.0f;

  // ---- prologue: stage tile 0 into buffer 0 ----
  {
    const int z0 = tile0 * 16;
#if HAVE_ASYNC_LDS
    stage_async(z0, 0);
#else
#pragma unroll
    for (int i = 0; i < 8; ++i) {
      int idx = tid + (i << 8);
      int m = idx >> 5, c = idx & 31;
      int zm = z0 + m; if (zm >= Z) zm = Z - 1;
      sreg[i] = *(const v4f*)(x1 + (size_t)zm * 512 + c * 4);
    }
#endif
    if (tid < 64) {
      int zm = z0 + (tid >> 2); if (zm >= Z) zm = Z - 1;
      x2v = x2[(size_t)zm * 4 + (tid & 3)];
    }
#if HAVE_ASYNC_LDS
    wait_async_all();
#else
#pragma unroll
    for (int i = 0; i < 8; ++i) {
      int idx = tid + (i << 8);
      int m = idx >> 5, c = idx & 31;
      *(v4f*)(smem + m * PX1 + c * 4) = sreg[i];
    }
#endif
    if (tid < 64) smem[OFF_X2 + tid] = x2v;
    __syncthreads();
  }

  for (int t = 0; t < TPB; ++t) {
    const int tile = tile0 + t;
    if (tile >= nTiles) break;            // uniform across block
    const int buf = t & 1, nbuf = buf ^ 1;
    const bool havenext = (t + 1 < TPB) && (tile + 1 < nTiles);

    // ---- issue next tile's copy; it proceeds concurrently with the WMMAs ----
    if (havenext) {
      const int z0n = (tile + 1) * 16;
#if HAVE_ASYNC_LDS
      stage_async(z0n, nbuf);
#else
#pragma unroll
      for (int i = 0; i < 8; ++i) {
        int idx = tid + (i << 8);
        int m = idx >> 5, c = idx & 31;
        int zm = z0n + m; if (zm >= Z) zm = Z - 1;
        sreg[i] = *(const v4f*)(x1 + (size_t)zm * 512 + c * 4);
      }
#endif
      if (tid < 64) {
        int zm = z0n + (tid >> 2); if (zm >= Z) zm = Z - 1;
        x2v = x2[(size_t)zm * 4 + (tid & 3)];
      }
    }

    // ---- compute: 6 accumulators over K=128 (32 x WMMA f32 16x16x4) ----
    const float* bx  = smem + buf * X1BUF;
    const float* bx2 = smem + OFF_X2 + buf * 64;
    const float q1 = bx2[nn * 4 + 1];
    const float q2 = bx2[nn * 4 + 2];
    const float q3 = bx2[nn * 4 + 3];

    v8f acc1 = {}, acc2 = {}, acc30 = {}, acc31 = {}, acc32 = {}, acc4 = {};
    const float* rowS = bx + nn * PX1 + 2 * h;            // x1 scalar part
    const float* rowV = bx + nn * PX1 + 128 + 6 * h;      // x1 vector part (3*u0)

#pragma unroll
    for (int kk = 0; kk < 32; ++kk) {
      v2f aS = *(const v2f*)(rowS + 4 * kk);
      const float* pv = rowV + 12 * kk;                   // 6 contiguous floats
      v2f a0 = mk2(pv[0], pv[3]);
      v2f a1 = mk2(pv[1], pv[4]);
      v2f a2 = mk2(pv[2], pv[5]);
      v2f aB = a0 * q1 + a1 * q2 + a2 * q3;               // packed FMAs
      acc1  = WMMA(aS, B0[kk], acc1);
      acc2  = WMMA(aS, B1[kk], acc2);
      acc30 = WMMA(a0, B2[kk], acc30);
      acc31 = WMMA(a1, B2[kk], acc31);
      acc32 = WMMA(a2, B2[kk], acc32);
      acc4  = WMMA(aB, B3[kk], acc4);
    }

    // ---- combine with per-z x2 scalars, store (non-temporal: streamed) ----
    // C/D layout: vgpr r, lane l -> M = r + 8*(l>>4), N = l&15
    const int z0 = tile * 16;
#pragma unroll
    for (int r = 0; r < 8; ++r) {
      int m = r + 8 * h;
      int zm = z0 + m;
      if (zm >= Z) continue;
      float s0 = bx2[m * 4 + 0];
      float sx = bx2[m * 4 + 1];
      float sy = bx2[m * 4 + 2];
      float sz = bx2[m * 4 + 3];
      size_t row = (size_t)zm * 512;
      __builtin_nontemporal_store(
          0.0625f * (s0 * acc1[r] + INV_SQRT3 * acc4[r]), &out[row + colbase + nn]);
      size_t bb = row + 128 + 3 * (size_t)(colbase + nn);
      __builtin_nontemporal_store(0.0625f * (sx * acc2[r] + s0 * acc30[r]), &out[bb + 0]);
      __builtin_nontemporal_store(0.0625f * (sy * acc2[r] + s0 * acc31[r]), &out[bb + 1]);
      __builtin_nontemporal_store(0.0625f * (sz * acc2[r] + s0 * acc32[r]), &out[bb + 2]);
    }

    // ---- complete next tile's staging, hand off buffers ----
    if (havenext) {
#if HAVE_ASYNC_LDS
      wait_async_all();
#else
      float* bn = smem + nbuf * X1BUF;
#pragma unroll
      for (int i = 0; i < 8; ++i) {
        int idx = tid + (i << 8);
        int m = idx >> 5, c = idx & 31;
        *(v4f*)(bn + m * PX1 + c * 4) = sreg[i];
      }
#endif
      if (tid < 64) smem[OFF_X2 + nbuf * 64 + tid] = x2v;
      __syncthreads();
    }
  }
}

extern "C" void kernel_launch(void* const* d_in, const int* in_sizes, int n_in,
                              void* d_out, int out_size, void* d_ws, size_t ws_size,
                              hipStream_t stream) {
  const float* x1 = (const float*)d_in[0];
  const float* x2 = (const float*)d_in[1];
  const float* wgt = (const float*)d_in[2];
  float* out = (float*)d_out;

  const int Z = in_sizes[0] / 512;
  const int nTiles = (Z + 15) / 16;
  const int blocks = (nTiles + TPB - 1) / TPB;
  const size_t shmem = (size_t)SMEM_FLOATS * sizeof(float);

  tp_uvw_wmma_kernel<<<blocks, 256, shmem, stream>>>(x1, x2, wgt, out, Z, nTiles);
}